// AnatomicalAttention_31662498906303
// MI455X (gfx1250) — compile-verified
//
#include <hip/hip_runtime.h>

typedef float v2f __attribute__((ext_vector_type(2)));
typedef float v8f __attribute__((ext_vector_type(8)));

#define FDIM 768
#define RDIM 7
#define BD   (4096 * 14)
#define WAVES 4

// ---------------- prep kernel 1: K[7x768], V[8x768] (row 7 zeroed) -------------
__global__ void aa_prep1(const float* __restrict__ regE,
                         const float* __restrict__ Wk, const float* __restrict__ bk,
                         const float* __restrict__ Wv, const float* __restrict__ bv,
                         float* __restrict__ Kmat, float* __restrict__ Vmat) {
    int idx = blockIdx.x * blockDim.x + threadIdx.x;
    if (blockIdx.y == 0) {
        if (idx >= RDIM * FDIM) return;
        int r = idx / FDIM, g = idx % FDIM;
        float acc = bk[g];
        const float* re = regE + r * FDIM;
        const float* w  = Wk + g * FDIM;
        for (int f = 0; f < FDIM; ++f) acc += re[f] * w[f];
        Kmat[idx] = acc;
    } else {
        if (idx >= 8 * FDIM) return;
        int r = idx / FDIM, g = idx % FDIM;
        if (r >= RDIM) { Vmat[idx] = 0.0f; return; }
        float acc = bv[g];
        const float* re = regE + r * FDIM;
        const float* w  = Wv + g * FDIM;
        for (int f = 0; f < FDIM; ++f) acc += re[f] * w[f];
        Vmat[idx] = acc;
    }
}

// ---- prep kernel 2: Kt2[r,f] = (K[r,:]@Wq)[f] * scale * rw[r] (8 rows, row7=0);
//                     c2[r]   = (bq . K[r,:]) * scale * rw[r]
__global__ void aa_prep2(const float* __restrict__ Kmat,
                         const float* __restrict__ Wq, const float* __restrict__ bq,
                         const float* __restrict__ rw,
                         float* __restrict__ Kt2, float* __restrict__ c2) {
    int idx = blockIdx.x * blockDim.x + threadIdx.x;
    const float scale = 1.0f / sqrtf((float)FDIM);
    if (idx < 8 * FDIM) {
        int r = idx / FDIM, f = idx % FDIM;
        if (r >= RDIM) {
            Kt2[idx] = 0.0f;
        } else {
            float acc = 0.0f;
            const float* kr = Kmat + r * FDIM;
            for (int g = 0; g < FDIM; ++g) acc += kr[g] * Wq[g * FDIM + f];
            Kt2[idx] = acc * scale * rw[r];
        }
    }
    if (idx < 8) {
        int r = idx;
        if (r >= RDIM) { c2[r] = 0.0f; }
        else {
            float acc = 0.0f;
            const float* kr = Kmat + r * FDIM;
            for (int g = 0; g < FDIM; ++g) acc += bq[g] * kr[g];
            c2[r] = acc * scale * rw[r];
        }
    }
}

// ---------------- main kernel: one 16-row tile per wave ------------------------
__global__ __launch_bounds__(32 * WAVES)
void aa_main(const float* __restrict__ feat,
             const float* __restrict__ Kt2g, const float* __restrict__ Vg,
             const float* __restrict__ c2g,
             float* __restrict__ outAtt, float* __restrict__ outAttn) {
    __shared__ __align__(16) float sKt[8 * FDIM];          // 24 KB (scaled K-tilde, row7=0)
    __shared__ __align__(16) float sV [8 * FDIM];          // 24 KB (V, row7=0)
    __shared__ __align__(16) float sAttn[WAVES][16][8];    // 2 KB

    const int tid = threadIdx.x;

    // cooperative stage of Kt2 and V into LDS (float4 coalesced)
    for (int i = tid; i < (8 * FDIM) / 4; i += blockDim.x) {
        ((float4*)sKt)[i] = ((const float4*)Kt2g)[i];
        ((float4*)sV )[i] = ((const float4*)Vg )[i];
    }
    __syncthreads();

    const int wave  = tid >> 5;
    const int lane  = tid & 31;
    const int row16 = lane & 15;   // M/N index within tile
    const int half  = lane >> 4;   // selects K pair {0,1} vs {2,3} in f32 frags
    const long rowBase = ((long)blockIdx.x * WAVES + wave) * 16;

    // ---------------- phase 1: scores^T = Kt2(A: 16x4) x feat^T(B: 4x16) -------
    // A: lane holds Kt2[region = row16 (pad rows -> zero row 7)][k + 2*half + {0,1}]
    // B: lane holds feat[row = row16][k + 2*half + {0,1}]
    const int regIdx = (row16 < 8) ? row16 : 7;
    const float* aPtr = sKt + regIdx * FDIM + 2 * half;
    const float* bPtr = feat + (rowBase + row16) * FDIM + 2 * half;

    v8f acc = {};
    #pragma unroll 4
    for (int k = 0; k < FDIM; k += 4) {
        v2f a = *(const v2f*)(aPtr + k);   // ds_load_b64
        v2f b = *(const v2f*)(bPtr + k);   // global_load_b64
        acc = __builtin_amdgcn_wmma_f32_16x16x4_f32(false, a, false, b,
                                                    (short)0, acc, false, false);
    }

    // ---------------- phase 2: per-lane softmax over 7 regions -----------------
    // D layout: lanes 0-15 hold regions 0..7 in acc[0..7] for row N=lane.
    if (lane < 16) {
        float s[RDIM];
        float m = -1e30f;
        #pragma unroll
        for (int r = 0; r < RDIM; ++r) { s[r] = acc[r] + c2g[r]; m = fmaxf(m, s[r]); }
        float sum = 0.0f;
        #pragma unroll
        for (int r = 0; r < RDIM; ++r) { s[r] = __expf(s[r] - m); sum += s[r]; }
        const float inv = 1.0f / sum;
        const long row = rowBase + lane;
        #pragma unroll
        for (int r = 0; r < RDIM; ++r) {
            const float a = s[r] * inv;
            sAttn[wave][lane][r] = a;
            outAttn[row * RDIM + r] = a;
        }
        sAttn[wave][lane][7] = 0.0f;
    }
    __builtin_amdgcn_wave_barrier();
    asm volatile("s_wait_dscnt 0" ::: "memory");   // cross-lane LDS visibility (same wave)

    // ---------------- phase 3: attended = attn(16x8) x V(8x16 chunks) ----------
    // A kstep0: attn[row16][2*half + {0,1}], kstep1: attn[row16][4 + 2*half + {0,1}]
    const v2f aLo = *(const v2f*)&sAttn[wave][row16][2 * half];
    const v2f aHi = *(const v2f*)&sAttn[wave][row16][4 + 2 * half];

    float* outBase = outAtt + rowBase * FDIM;
    for (int fb = 0; fb < FDIM; fb += 16) {
        const int col = fb + row16;
        v2f b0, b1;
        b0.x = sV[(2 * half + 0) * FDIM + col];
        b0.y = sV[(2 * half + 1) * FDIM + col];
        b1.x = sV[(4 + 2 * half) * FDIM + col];
        b1.y = sV[(5 + 2 * half) * FDIM + col];
        v8f d = {};
        d = __builtin_amdgcn_wmma_f32_16x16x4_f32(false, aLo, false, b0,
                                                  (short)0, d, false, false);
        d = __builtin_amdgcn_wmma_f32_16x16x4_f32(false, aHi, false, b1,
                                                  (short)0, d, false, false);
        // D: VGPR v -> row v + 8*half, col = fb + row16 (64B contiguous per half-wave)
        #pragma unroll
        for (int v = 0; v < 8; ++v) {
            outBase[(long)(v + 8 * half) * FDIM + col] = d[v];
        }
    }
}

// ---------------------------------------------------------------------------
extern "C" void kernel_launch(void* const* d_in, const int* in_sizes, int n_in,
                              void* d_out, int out_size, void* d_ws, size_t ws_size,
                              hipStream_t stream) {
    (void)in_sizes; (void)n_in; (void)out_size; (void)ws_size;
    const float* feat = (const float*)d_in[0];
    const float* regE = (const float*)d_in[1];
    const float* Wq   = (const float*)d_in[2];
    const float* bq   = (const float*)d_in[3];
    const float* Wk   = (const float*)d_in[4];
    const float* bk   = (const float*)d_in[5];
    const float* Wv   = (const float*)d_in[6];
    const float* bv   = (const float*)d_in[7];
    const float* rw   = (const float*)d_in[8];

    float* ws   = (float*)d_ws;
    float* Kmat = ws;                       // 7*768
    float* Vmat = Kmat + RDIM * FDIM;       // 8*768
    float* Kt2  = Vmat + 8 * FDIM;          // 8*768
    float* c2   = Kt2 + 8 * FDIM;           // 8

    float* outAtt  = (float*)d_out;
    float* outAttn = outAtt + (long)BD * FDIM;

    dim3 g1((8 * FDIM + 255) / 256, 2);
    aa_prep1<<<g1, 256, 0, stream>>>(regE, Wk, bk, Wv, bv, Kmat, Vmat);
    aa_prep2<<<(8 * FDIM + 255) / 256, 256, 0, stream>>>(Kmat, Wq, bq, rw, Kt2, c2);

    const int tiles  = BD / 16;            // 3584
    const int blocks = tiles / WAVES;      // 896
    aa_main<<<blocks, 32 * WAVES, 0, stream>>>(feat, Kt2, Vmat, c2, outAtt, outAttn);
}